// InfoAggregator_5617817223548
// MI455X (gfx1250) — compile-verified
//
#include <hip/hip_runtime.h>

#define NN 2048   // nodes
#define DD 256    // embedding dim
#define CC 768    // 3*DD
#define VV 2      // variants
#define LL 16     // activities per variant
#define EE (16 * NN) // edges
#define KTOT (2 * DD) // fused GEMM K = [emb | msg]

typedef float v2f __attribute__((ext_vector_type(2)));
typedef float v8f __attribute__((ext_vector_type(8)));

// ---------------------------------------------------------------------------
// Init: copy embeddings into per-variant working buffer, zero pi and summary.
// ---------------------------------------------------------------------------
__global__ __launch_bounds__(256) void init_state_kernel(
    const float* __restrict__ embeddings, float* __restrict__ embA,
    float* __restrict__ pi, float* __restrict__ summary)
{
    const size_t i = (size_t)blockIdx.x * 256 + threadIdx.x;
    if (i < (size_t)VV * NN * DD) embA[i] = embeddings[i % ((size_t)NN * DD)];
    if (i < (size_t)VV * NN * CC) pi[i] = 0.f;
    if (i < (size_t)VV * CC) summary[i] = 0.f;
}

// ---------------------------------------------------------------------------
// Deterministic CSR-by-destination build (edges are static across steps).
// ---------------------------------------------------------------------------
__global__ __launch_bounds__(256) void csr_count_kernel(
    const int* __restrict__ dst_e, int* __restrict__ cnt)
{
    const int d = blockIdx.x * 256 + threadIdx.x;
    if (d >= NN) return;
    int c = 0;
    for (int e = 0; e < EE; ++e) c += (dst_e[e] == d) ? 1 : 0;
    cnt[d] = c;
}

__global__ void csr_scan_kernel(const int* __restrict__ cnt, int* __restrict__ rowPtr)
{
    if (threadIdx.x == 0 && blockIdx.x == 0) {
        int s = 0;
        for (int d = 0; d < NN; ++d) { rowPtr[d] = s; s += cnt[d]; }
        rowPtr[NN] = s;
    }
}

__global__ __launch_bounds__(256) void csr_fill_kernel(
    const int* __restrict__ src_e, const int* __restrict__ dst_e,
    const int* __restrict__ rowPtr, int* __restrict__ colIdx)
{
    const int d = blockIdx.x * 256 + threadIdx.x;
    if (d >= NN) return;
    int pos = rowPtr[d];
    for (int e = 0; e < EE; ++e)
        if (dst_e[e] == d) colIdx[pos++] = src_e[e]; // edge order: deterministic fp sums
}

// ---------------------------------------------------------------------------
// Per-step places/summary bookkeeping (tiny, one workgroup per variant).
// mask is nonzero only on followers of `a` that reach a later-in-variant node.
// ---------------------------------------------------------------------------
__global__ __launch_bounds__(256) void places_step_kernel(
    const float* __restrict__ embBase, float* __restrict__ piBase,
    float* __restrict__ sumBase, const int* __restrict__ adj,
    const int* __restrict__ variants, int step, int* __restrict__ hasFlag)
{
    const int v = blockIdx.y;
    const float* emb = embBase + (size_t)v * NN * DD;
    float* pi = piBase + (size_t)v * NN * CC;
    float* sumG = sumBase + (size_t)v * CC;

    __shared__ unsigned char isM[NN];
    __shared__ unsigned short conn[NN];
    __shared__ int later[LL];
    __shared__ int nlaterS;
    __shared__ int anyF;
    __shared__ float sumL[CC];

    const int tid = threadIdx.x;
    if (tid == 0) {
        anyF = 0;
        int nl = 0;
        for (int j = step + 1; j < LL; ++j) {      // distinct nodes appearing later
            const int id = variants[v * LL + j];
            bool dup = false;
            for (int t = 0; t < nl; ++t) if (later[t] == id) dup = true;
            if (!dup) later[nl++] = id;
        }
        nlaterS = nl;
    }
    for (int c = tid; c < CC; c += 256) sumL[c] = sumG[c];
    __syncthreads();

    const int a = variants[v * LL + step];
    const int nl = nlaterS;

    for (int p = tid; p < NN; p += 256) {
        unsigned char msk = 0;
        unsigned short cn = 0;
        if (adj[(size_t)a * NN + p] != 0) {         // follower of a
            anyF = 1;                               // benign same-value race
            int dc = 0;
            for (int t = 0; t < nl; ++t)
                if (adj[(size_t)p * NN + later[t]] != 0) { cn |= (unsigned short)(1u << t); ++dc; }
            if (dc > 0) msk = 1;                    // dest_cnt > 0
        }
        isM[p] = msk;
        conn[p] = cn;
    }
    __syncthreads();
    if (tid == 0) hasFlag[v] = anyF;

    // Phase A: pi[masked] += cat = [emb[p], emb[a], dst_all[p]]
    for (int p = 0; p < NN; ++p) {
        if (!isM[p]) continue;
        const unsigned cn = conn[p];
        for (int c = tid; c < CC; c += 256) {
            float val;
            if (c < DD) val = emb[(size_t)p * DD + c];
            else if (c < 2 * DD) val = emb[(size_t)a * DD + (c - DD)];
            else {
                const int cc = c - 2 * DD;
                float s = 0.f;
                for (int t = 0; t < nl; ++t)
                    if (cn & (1u << t)) s += emb[(size_t)later[t] * DD + cc];
                val = s;
            }
            pi[(size_t)p * CC + c] += val;
        }
    }
    __syncthreads();

    // Phase B: summary += sum over masked rows of (updated) pi
    for (int c = tid; c < CC; c += 256) {
        float acc = sumL[c];
        for (int p = 0; p < NN; ++p)
            if (isM[p]) acc += pi[(size_t)p * CC + c];
        sumL[c] = acc;
    }
    __syncthreads();

    // Phase C: pi[masked] += new summary; write summary back
    for (int p = 0; p < NN; ++p) {
        if (!isM[p]) continue;
        for (int c = tid; c < CC; c += 256)
            pi[(size_t)p * CC + c] += sumL[c];
    }
    for (int c = tid; c < CC; c += 256) sumG[c] = sumL[c];
}

// ---------------------------------------------------------------------------
// msg[dst] = sum over incoming edges of emb[src]. One wave per dst row,
// lane-coalesced over the 256-float row; CSR order -> deterministic.
// Prefetch the next source row (global_prefetch_b8) while summing this one.
// ---------------------------------------------------------------------------
__global__ __launch_bounds__(256) void msg_kernel(
    const float* __restrict__ embBase,
    const int* __restrict__ rowPtr, const int* __restrict__ colIdx,
    float* __restrict__ msgBase)
{
    const int v = blockIdx.y;
    const float* emb = embBase + (size_t)v * NN * DD;
    float* msg = msgBase + (size_t)v * NN * DD;

    const int wave = threadIdx.x >> 5;
    const int lane = threadIdx.x & 31;
    const int dst = blockIdx.x * 8 + wave;

    float acc[8];
#pragma unroll
    for (int q = 0; q < 8; ++q) acc[q] = 0.f;

    const int beg = rowPtr[dst];
    const int end = rowPtr[dst + 1];
    for (int e = beg; e < end; ++e) {
        const float* src = emb + (size_t)colIdx[e] * DD;
        if (e + 1 < end)  // hide gather latency for the next indirect row
            __builtin_prefetch(emb + (size_t)colIdx[e + 1] * DD + lane * 8, 0, 3);
#pragma unroll
        for (int q = 0; q < 8; ++q) acc[q] += src[lane + q * 32];
    }
#pragma unroll
    for (int q = 0; q < 8; ++q) msg[(size_t)dst * DD + lane + q * 32] = acc[q];
}

// ---------------------------------------------------------------------------
// Fused encoder GEMM: out = relu([emb|msg] @ [[W_self],[W_nbr]] + b), or a
// copy of emb when the step has no followers (hasFlag==0).
// fp32 WMMA 16x16x4; A panel (16 x 512) staged in LDS with +4 padding so
// fragment reads hit 32 distinct banks. 8 waves x 2 col-tiles = 256 cols.
// K-loop split into two branch-free halves (W_self then W_nbr) so the hot
// loop is ds_load_b64 + coalesced W loads + 2 WMMA with no control flow.
// ---------------------------------------------------------------------------
__global__ __launch_bounds__(256) void encoder_gemm_kernel(
    const float* __restrict__ embCur, const float* __restrict__ msgBase,
    const float* __restrict__ Wself, const float* __restrict__ Wnbr,
    const float* __restrict__ bias, const int* __restrict__ hasFlag,
    float* __restrict__ embNext)
{
    const int v = blockIdx.y;
    const float* A0 = embCur + (size_t)v * NN * DD;
    const float* M0 = msgBase + (size_t)v * NN * DD;
    float* Out = embNext + (size_t)v * NN * DD;
    const int r0 = blockIdx.x << 4;

    __shared__ float lA[16][KTOT + 4]; // 516-float row stride: conflict-free frags

    const int tid = threadIdx.x;
    for (int idx = tid; idx < 16 * KTOT; idx += 256) {
        const int r = idx >> 9;           // / KTOT
        const int k = idx & (KTOT - 1);
        lA[r][k] = (k < DD) ? A0[(size_t)(r0 + r) * DD + k]
                            : M0[(size_t)(r0 + r) * DD + (k - DD)];
    }
    __syncthreads();

    const int wave = tid >> 5;
    const int lane = tid & 31;
    const int m = lane & 15;
    const int koff = (lane >> 4) << 1; // 0 for lanes 0-15, 2 for lanes 16-31
    const int c0 = wave << 5;          // tile 0 col base
    const int c1 = c0 + 16;            // tile 1 col base

    v8f acc0 = {0.f, 0.f, 0.f, 0.f, 0.f, 0.f, 0.f, 0.f};
    v8f acc1 = {0.f, 0.f, 0.f, 0.f, 0.f, 0.f, 0.f, 0.f};

    // ---- half 0: k in [0, DD), A = emb panel, B = W_self -------------------
    {
        const float* wp0 = Wself + (size_t)koff * DD + c0 + m;        // row k0
        const float* wp1 = Wself + (size_t)(koff + 1) * DD + c0 + m;  // row k0+1
        const float* ap  = &lA[m][koff];
#pragma unroll 8
        for (int kk = 0; kk < DD; kk += 4) {
            v2f a;
            a.x = ap[0];
            a.y = ap[1];
            v2f b0, b1;
            b0.x = wp0[0];  b0.y = wp1[0];
            b1.x = wp0[16]; b1.y = wp1[16];
            acc0 = __builtin_amdgcn_wmma_f32_16x16x4_f32(false, a, false, b0,
                                                         (short)0, acc0, false, false);
            acc1 = __builtin_amdgcn_wmma_f32_16x16x4_f32(false, a, false, b1,
                                                         (short)0, acc1, false, false);
            ap  += 4;
            wp0 += 4 * DD;
            wp1 += 4 * DD;
        }
    }
    // ---- half 1: k in [DD, 2*DD), A = msg panel, B = W_nbr -----------------
    {
        const float* wp0 = Wnbr + (size_t)koff * DD + c0 + m;
        const float* wp1 = Wnbr + (size_t)(koff + 1) * DD + c0 + m;
        const float* ap  = &lA[m][DD + koff];
#pragma unroll 8
        for (int kk = 0; kk < DD; kk += 4) {
            v2f a;
            a.x = ap[0];
            a.y = ap[1];
            v2f b0, b1;
            b0.x = wp0[0];  b0.y = wp1[0];
            b1.x = wp0[16]; b1.y = wp1[16];
            acc0 = __builtin_amdgcn_wmma_f32_16x16x4_f32(false, a, false, b0,
                                                         (short)0, acc0, false, false);
            acc1 = __builtin_amdgcn_wmma_f32_16x16x4_f32(false, a, false, b1,
                                                         (short)0, acc1, false, false);
            ap  += 4;
            wp0 += 4 * DD;
            wp1 += 4 * DD;
        }
    }

    const int has = hasFlag[v];
    const int half = (lane >> 4) << 3; // C/D layout: lanes 16-31 hold rows M+8
    const int ca = c0 + m;
    const int cb = c1 + m;
    const float bA = bias[ca];
    const float bB = bias[cb];
#pragma unroll
    for (int g = 0; g < 8; ++g) {
        const int rr = g + half;
        const int row = r0 + rr;
        const float x0 = fmaxf(acc0[g] + bA, 0.f);
        const float x1 = fmaxf(acc1[g] + bB, 0.f);
        Out[(size_t)row * DD + ca] = has ? x0 : lA[rr][ca];
        Out[(size_t)row * DD + cb] = has ? x1 : lA[rr][cb];
    }
}

// ---------------------------------------------------------------------------
// Output: sum pi over the two variants.
// ---------------------------------------------------------------------------
__global__ __launch_bounds__(256) void final_sum_kernel(
    const float* __restrict__ pi, float* __restrict__ out)
{
    const size_t i = (size_t)blockIdx.x * 256 + threadIdx.x;
    if (i < (size_t)NN * CC) out[i] = pi[i] + pi[i + (size_t)NN * CC];
}

// ---------------------------------------------------------------------------
extern "C" void kernel_launch(void* const* d_in, const int* in_sizes, int n_in,
                              void* d_out, int out_size, void* d_ws, size_t ws_size,
                              hipStream_t stream)
{
    (void)in_sizes; (void)n_in; (void)out_size; (void)ws_size;
    const float* embeddings = (const float*)d_in[0];
    const float* W_self     = (const float*)d_in[1];
    const float* W_nbr      = (const float*)d_in[2];
    const float* bias       = (const float*)d_in[3];
    const int*   variants   = (const int*)d_in[4];
    const int*   adj        = (const int*)d_in[5];
    const int*   edge_index = (const int*)d_in[6];
    const int*   src_e = edge_index;
    const int*   dst_e = edge_index + EE;

    char* ws = (char*)d_ws;
    size_t off = 0;
    auto carve = [&](size_t bytes) -> void* {
        void* p = ws + off;
        off = (off + bytes + 255) & ~(size_t)255;
        return p;
    };
    float* embA    = (float*)carve(sizeof(float) * VV * NN * DD);
    float* embB    = (float*)carve(sizeof(float) * VV * NN * DD);
    float* msg     = (float*)carve(sizeof(float) * VV * NN * DD);
    float* pi      = (float*)carve(sizeof(float) * VV * NN * CC);
    float* summary = (float*)carve(sizeof(float) * VV * CC);
    int*   rowPtr  = (int*)carve(sizeof(int) * (NN + 1));
    int*   colIdx  = (int*)carve(sizeof(int) * EE);
    int*   cnt     = (int*)carve(sizeof(int) * NN);
    int*   hasFlag = (int*)carve(sizeof(int) * VV);

    init_state_kernel<<<dim3((VV * NN * CC + 255) / 256), 256, 0, stream>>>(
        embeddings, embA, pi, summary);
    csr_count_kernel<<<dim3((NN + 255) / 256), 256, 0, stream>>>(dst_e, cnt);
    csr_scan_kernel<<<1, 64, 0, stream>>>(cnt, rowPtr);
    csr_fill_kernel<<<dim3((NN + 255) / 256), 256, 0, stream>>>(src_e, dst_e, rowPtr, colIdx);

    float* cur = embA;
    float* nxt = embB;
    for (int step = 0; step < LL; ++step) {
        places_step_kernel<<<dim3(1, VV), 256, 0, stream>>>(
            cur, pi, summary, adj, variants, step, hasFlag);
        msg_kernel<<<dim3(NN / 8, VV), 256, 0, stream>>>(cur, rowPtr, colIdx, msg);
        encoder_gemm_kernel<<<dim3(NN / 16, VV), 256, 0, stream>>>(
            cur, msg, W_self, W_nbr, bias, hasFlag, nxt);
        float* t = cur; cur = nxt; nxt = t;
    }
    final_sum_kernel<<<dim3((NN * CC + 255) / 256), 256, 0, stream>>>(pi, (float*)d_out);
}